// FCOS_26362509262898
// MI455X (gfx1250) — compile-verified
//
#include <hip/hip_runtime.h>
#include <stdint.h>

// ---------------- constants (match reference) ----------------
#define NC        80
#define NLEV      5
#define NIMG      16
#define PRE       1000         // NMS_PRE_TOPK per level
#define NREAL     (NLEV*PRE)   // 5000 real candidates / image
#define CANDP     5120         // padded to multiple of 16
#define POST      100          // NMS_POST_TOPK
#define NBIN      4096         // histogram bins for top-k threshold
#define CAP       4096         // per (img,level) compaction capacity
#define SCORE_TH  0.05f
#define NMS_TH    0.6f
#define CLS_OFF   10000.0f
#define CSTRIDE   8            // AoS stride: score, cls, x1, y1, x2, y2, pad, pad

typedef float v2f __attribute__((ext_vector_type(2)));
typedef float v8f __attribute__((ext_vector_type(8)));
typedef unsigned long long u64;
typedef unsigned u32;

__device__ __forceinline__ float sigmoidf(float x) { return 1.0f / (1.0f + __expf(-x)); }

// order-preserving float<->u32 map (sortable as unsigned, ascending)
__device__ __forceinline__ u32 fmap(float f) {
  u32 u = __float_as_uint(f);
  return (u & 0x80000000u) ? ~u : (u | 0x80000000u);
}
__device__ __forceinline__ float funmap(u32 u) {
  u32 v = (u & 0x80000000u) ? (u & 0x7FFFFFFFu) : ~u;
  return __uint_as_float(v);
}

// ---------------- tiny utility kernels ----------------
__global__ void k_zero_u32(u32* __restrict__ p, int n) {
  int i = blockIdx.x * blockDim.x + threadIdx.x;
  if (i < n) p[i] = 0u;
}

__global__ void k_ctr_sigmoid(const float* __restrict__ ctr, float* __restrict__ out, int n) {
  int i = blockIdx.x * blockDim.x + threadIdx.x;
  if (i < n) out[i] = sigmoidf(ctr[i]);
}

// ---------------- pass 1: per (img,level) score histogram ----------------
// grid: (ceil(P/256), NIMG*NC); blockIdx.y -> (img, class) with compile-time /80
__global__ void k_hist(const float* __restrict__ cls, const float* __restrict__ ctrS,
                       u32* __restrict__ hist, int P, int lev) {
  int p = blockIdx.x * blockDim.x + threadIdx.x;
  if (p >= P) return;
  int img = blockIdx.y / NC;
  int c = blockIdx.y % NC;
  float s = sigmoidf(cls[(size_t)img * NC * P + (size_t)c * P + p]);
  if (s > SCORE_TH) {
    float v = s * ctrS[img * P + p];
    int bin = (int)(v * (float)NBIN);
    bin = bin < 0 ? 0 : (bin > NBIN - 1 ? NBIN - 1 : bin);
    atomicAdd(&hist[(size_t)(img * NLEV + lev) * NBIN + bin], 1u);
  }
}

// pick bin threshold so that count(bins >= t) >= PRE (exact top-k lives in bins >= t)
__global__ void k_thresh(const u32* __restrict__ hist, int* __restrict__ thr) {
  int g = blockIdx.x * blockDim.x + threadIdx.x;
  if (g >= NIMG * NLEV) return;
  const u32* h = hist + (size_t)g * NBIN;
  u32 cum = 0;
  int t = 0;
  for (int b = NBIN - 1; b >= 0; --b) {
    cum += h[b];
    if (cum >= PRE) { t = b; break; }
  }
  thr[g] = t;
}

// ---------------- pass 2: compact survivors as sortable u64 keys ----------------
__global__ void k_compact(const float* __restrict__ cls, const float* __restrict__ ctrS,
                          const int* __restrict__ thr, u32* __restrict__ cnt,
                          u64* __restrict__ keys, int P, int lev) {
  int p = blockIdx.x * blockDim.x + threadIdx.x;
  if (p >= P) return;
  int img = blockIdx.y / NC;
  int c = blockIdx.y % NC;
  float s = sigmoidf(cls[(size_t)img * NC * P + (size_t)c * P + p]);
  if (s > SCORE_TH) {
    float v = s * ctrS[img * P + p];
    int bin = (int)(v * (float)NBIN);
    bin = bin < 0 ? 0 : (bin > NBIN - 1 ? NBIN - 1 : bin);
    int g = img * NLEV + lev;
    if (bin >= thr[g]) {
      u32 pos = atomicAdd(&cnt[g], 1u);
      if (pos < CAP) {
        u32 idx = (u32)(p * NC + c);  // reference: idx = p*C + c
        // score descending, index ascending tie-break (matches jax.lax.top_k)
        keys[(size_t)g * CAP + pos] = ((u64)fmap(v) << 32) | (u64)(0xFFFFFFFFu - idx);
      }
    }
  }
}

// ---------------- per (img,level): sort survivors, emit exact top-1000 boxes ----------------
struct LevelMeta {
  const float* box[NLEV];
  int P[NLEV], W[NLEV], S[NLEV];
};

__global__ void __launch_bounds__(512)
k_level_topk(const u32* __restrict__ cnt, const u64* __restrict__ keys,
             LevelMeta L, float* __restrict__ cand) {
  __shared__ u64 sh[CAP];  // 32 KB
  int g = blockIdx.x;
  int img = g / NLEV, lev = g % NLEV;
  int n = (int)cnt[g];
  if (n > CAP) n = CAP;
  for (int k = threadIdx.x; k < CAP; k += blockDim.x)
    sh[k] = (k < n) ? keys[(size_t)g * CAP + k] : 0ULL;
  __syncthreads();
  // bitonic sort, descending
  for (int k2 = 2; k2 <= CAP; k2 <<= 1) {
    for (int j = k2 >> 1; j > 0; j >>= 1) {
      for (int i = threadIdx.x; i < CAP; i += blockDim.x) {
        int ixj = i ^ j;
        if (ixj > i) {
          u64 a = sh[i], b = sh[ixj];
          bool descRegion = ((i & k2) == 0);
          if (descRegion ? (a < b) : (a > b)) { sh[i] = b; sh[ixj] = a; }
        }
      }
      __syncthreads();
    }
  }
  int P = L.P[lev], W = L.W[lev], S = L.S[lev];
  const float* box = L.box[lev] + (size_t)img * 4 * P;
  for (int rank = threadIdx.x; rank < PRE; rank += blockDim.x) {
    size_t o = ((size_t)img * CANDP + (size_t)lev * PRE + rank) * CSTRIDE;
    float4 f0, f1;
    if (rank < n) {
      u64 key = sh[rank];
      float v = funmap((u32)(key >> 32));
      u32 idx = 0xFFFFFFFFu - (u32)(key & 0xFFFFFFFFULL);
      int p = (int)(idx / NC), c = (int)(idx % NC);
      float px = (float)((p % W) * S + (S >> 1));
      float py = (float)((p / W) * S + (S >> 1));
      float fs = (float)S;
      float r0 = box[0 * P + p] * fs, r1 = box[1 * P + p] * fs;
      float r2 = box[2 * P + p] * fs, r3 = box[3 * P + p] * fs;
      f0 = make_float4(sqrtf(v), (float)c, px - r0, py - r1);
      f1 = make_float4(px + r2, py + r3, 0.0f, 0.0f);
    } else {  // below-threshold pad: invalid candidate
      f0 = make_float4(-1.0f, -1.0f, 0.0f, 0.0f);
      f1 = make_float4(0.0f, 0.0f, 0.0f, 0.0f);
    }
    *(float4*)&cand[o + 0] = f0;
    *(float4*)&cand[o + 4] = f1;
  }
}

// ---------------- per image: sort 5120 candidates by (score desc, pos asc) ----------------
__global__ void __launch_bounds__(1024)
k_image_sort(const float* __restrict__ cand, float* __restrict__ sorted) {
  __shared__ u64 sh[8192];  // 64 KB
  int img = blockIdx.x;
  for (int k = threadIdx.x; k < 8192; k += blockDim.x) {
    u64 key = 0ULL;
    if (k < CANDP) {
      float scv = (k < NREAL) ? cand[((size_t)img * CANDP + k) * CSTRIDE] : -1.0f;
      key = ((u64)fmap(scv) << 32) | (u64)(0xFFFFFFFFu - (u32)k);
    }
    sh[k] = key;
  }
  __syncthreads();
  for (int k2 = 2; k2 <= 8192; k2 <<= 1) {
    for (int j = k2 >> 1; j > 0; j >>= 1) {
      for (int i = threadIdx.x; i < 8192; i += blockDim.x) {
        int ixj = i ^ j;
        if (ixj > i) {
          u64 a = sh[i], b = sh[ixj];
          bool descRegion = ((i & k2) == 0);
          if (descRegion ? (a < b) : (a > b)) { sh[i] = b; sh[ixj] = a; }
        }
      }
      __syncthreads();
    }
  }
  for (int r = threadIdx.x; r < CANDP; r += blockDim.x) {
    u64 key = sh[r];
    u32 pos = 0xFFFFFFFFu - (u32)(key & 0xFFFFFFFFULL);
    size_t o = ((size_t)img * CANDP + r) * CSTRIDE;
    float4 f0, f1;
    if (key != 0ULL && pos < (u32)NREAL) {
      size_t q = ((size_t)img * CANDP + pos) * CSTRIDE;
      f0 = *(const float4*)&cand[q + 0];
      f1 = *(const float4*)&cand[q + 4];
    } else {
      f0 = make_float4(-1.0f, -1.0f, 0.0f, 0.0f);
      f1 = make_float4(0.0f, 0.0f, 0.0f, 0.0f);
    }
    *(float4*)&sorted[o + 0] = f0;
    *(float4*)&sorted[o + 4] = f1;
  }
}

// ---------------- per image: blocked greedy class-aware NMS + top-100 output ----------------
// 16-anchor blocks: intra-block dependency resolved serially by thread 0, block-vs-rest
// suppression is 16xN matrix work. Per 16x16 tile the IoU denominator (area_i + area_j) is a
// rank-2 outer product computed on the matrix unit via V_WMMA_F32_16X16X4_F32.
// iou > thr is evaluated division-free and branchlessly: denom>0 && inter > thr*denom,
// gated by a per-block anchor-keep bitmask (no EXEC churn in the hot loop).
__global__ void __launch_bounds__(1024)
k_nms(const float* __restrict__ sorted, float* __restrict__ out) {
  __shared__ u32 keep[CANDP];  // 20 KB
  __shared__ float ax1[16], ay1[16], ax2[16], ay2[16], aarea[16];
  __shared__ u32 akeep[16];
  __shared__ u32 amaskSh;
  int img = blockIdx.x;
  const float* cbuf = sorted + (size_t)img * CANDP * CSTRIDE;

  for (int r = threadIdx.x; r < CANDP; r += blockDim.x)
    keep[r] = (cbuf[(size_t)r * CSTRIDE] >= 0.0f) ? 1u : 0u;  // valid flags seed keep
  __syncthreads();

  const int wid = threadIdx.x >> 5;   // wave id, 0..31 (wave32)
  const int lane = threadIdx.x & 31;
  const int halfId = lane >> 4;       // 0: C/D rows 0..7, 1: rows 8..15 (WMMA layout)
  const int ln = lane & 15;           // N column within tile

  for (int base = 0; base < CANDP; base += 16) {
    if (threadIdx.x == 0) {
      // load anchor block (class-offset boxes) + serial intra-block greedy
      for (int i = 0; i < 16; ++i) {
        const float* cp = cbuf + (size_t)(base + i) * CSTRIDE;
        float off = cp[1] * CLS_OFF;
        float x1 = cp[2] + off, y1 = cp[3] + off, x2 = cp[4] + off, y2 = cp[5] + off;
        ax1[i] = x1; ay1[i] = y1; ax2[i] = x2; ay2[i] = y2;
        aarea[i] = fmaxf(x2 - x1, 0.0f) * fmaxf(y2 - y1, 0.0f);
        akeep[i] = keep[base + i];
      }
      for (int i = 0; i < 16; ++i) {
        if (!akeep[i]) continue;
        for (int jj = i + 1; jj < 16; ++jj) {
          if (!akeep[jj]) continue;
          float iw = fminf(ax2[i], ax2[jj]) - fmaxf(ax1[i], ax1[jj]);
          float ih = fminf(ay2[i], ay2[jj]) - fmaxf(ay1[i], ay1[jj]);
          float inter = fmaxf(iw, 0.0f) * fmaxf(ih, 0.0f);
          float denom = aarea[i] + aarea[jj] - inter;
          if (denom > 0.0f && inter > NMS_TH * denom) akeep[jj] = 0u;
        }
      }
      u32 m = 0u;
      for (int i = 0; i < 16; ++i) {
        keep[base + i] = akeep[i];
        m |= akeep[i] ? (1u << i) : 0u;
      }
      amaskSh = m;
    }
    __syncthreads();

    u32 amask = amaskSh;  // block-uniform
    if (amask != 0u) {    // uniform branch: EXEC stays all-ones for WMMA
      // hoist tile-invariant anchor state into registers (this lane's 8 C/D rows)
      float Aax1[8], Aay1[8], Aax2[8], Aay2[8], Aarea[8];
#pragma unroll
      for (int r = 0; r < 8; ++r) {
        int m8 = r + (halfId << 3);
        Aax1[r] = ax1[m8]; Aay1[r] = ay1[m8];
        Aax2[r] = ax2[m8]; Aay2[r] = ay2[m8];
        Aarea[r] = aarea[m8];
      }
      v2f va;
      va.x = halfId ? 0.0f : aarea[ln];  // A col K=0 (K=2 half is zero)
      va.y = halfId ? 0.0f : 1.0f;       // A col K=1 (K=3 half is zero)

      int firstTile = (base >> 4) + 1;
      int nTiles = (CANDP >> 4) - firstTile;
      for (int t = wid; t < nTiles; t += 32) {  // wave-uniform loop
        int cb = (firstTile + t) << 4;
        if (t + 32 < nTiles)
          __builtin_prefetch(&cbuf[(size_t)(cb + (32 << 4)) * CSTRIDE], 0, 0);
        int j = cb + ln;
        const float* cp = cbuf + (size_t)j * CSTRIDE;
        float4 c0 = *(const float4*)&cp[0];  // score, cls, x1, y1
        float4 c1 = *(const float4*)&cp[4];  // x2, y2, -, -
        float off = c0.y * CLS_OFF;
        float cx1 = c0.z + off, cy1 = c0.w + off;
        float cx2 = c1.x + off, cy2 = c1.y + off;
        float carea = fmaxf(cx2 - cx1, 0.0f) * fmaxf(cy2 - cy1, 0.0f);
        v2f vb;
        vb.x = halfId ? 0.0f : 1.0f;   // B row K=0
        vb.y = halfId ? 0.0f : carea;  // B row K=1
        v8f acc = {0.0f, 0.0f, 0.0f, 0.0f, 0.0f, 0.0f, 0.0f, 0.0f};
#if defined(__has_builtin) && __has_builtin(__builtin_amdgcn_wmma_f32_16x16x4_f32)
        acc = __builtin_amdgcn_wmma_f32_16x16x4_f32(false, va, false, vb, (short)0, acc,
                                                    false, false);
#else
#pragma unroll
        for (int r = 0; r < 8; ++r) acc[r] = Aarea[r] + carea;
#endif
        u32 sup = 0u;
#pragma unroll
        for (int r = 0; r < 8; ++r) {  // branchless, division-free IoU test
          float iw = fminf(Aax2[r], cx2) - fmaxf(Aax1[r], cx1);
          float ih = fminf(Aay2[r], cy2) - fmaxf(Aay1[r], cy1);
          float inter = fmaxf(iw, 0.0f) * fmaxf(ih, 0.0f);
          float denom = acc[r] - inter;  // area_m + area_n - inter (from WMMA)
          u32 hit = (denom > 0.0f && inter > NMS_TH * denom) ? 1u : 0u;
          sup |= hit & ((amask >> (r + (halfId << 3))) & 1u);
        }
        if (sup) keep[j] = 0u;  // both half-lanes may store 0: benign
      }
    }
    __syncthreads();
  }

  // outputs: boxes[16,100,4] | scores[16,100] | classes[16,100] | kept[16,100], flat floats
  if (threadIdx.x == 0) {
    float* ob = out;
    float* os = out + (size_t)NIMG * POST * 4;
    float* oc = os + (size_t)NIMG * POST;
    float* ok = oc + (size_t)NIMG * POST;
    int c2 = 0;
    for (int r = 0; r < CANDP && c2 < POST; ++r) {
      if (keep[r]) {
        const float* cp = cbuf + (size_t)r * CSTRIDE;
        size_t bo = ((size_t)img * POST + c2) * 4;
        ob[bo + 0] = cp[2]; ob[bo + 1] = cp[3]; ob[bo + 2] = cp[4]; ob[bo + 3] = cp[5];
        os[img * POST + c2] = cp[0];
        oc[img * POST + c2] = cp[1];
        ok[img * POST + c2] = 1.0f;
        ++c2;
      }
    }
    for (; c2 < POST; ++c2) {
      size_t bo = ((size_t)img * POST + c2) * 4;
      ob[bo + 0] = ob[bo + 1] = ob[bo + 2] = ob[bo + 3] = 0.0f;
      os[img * POST + c2] = -1.0f;
      oc[img * POST + c2] = -1.0f;
      ok[img * POST + c2] = 0.0f;
    }
  }
}

// ---------------- host launcher ----------------
extern "C" void kernel_launch(void* const* d_in, const int* in_sizes, int n_in,
                              void* d_out, int out_size, void* d_ws, size_t ws_size,
                              hipStream_t stream) {
  (void)in_sizes; (void)n_in; (void)out_size; (void)ws_size;
  static const int LP[NLEV] = {15200, 3800, 950, 247, 70};
  static const int LW_[NLEV] = {152, 76, 38, 19, 10};
  static const int LS_[NLEV] = {8, 16, 32, 64, 128};

  const float* clsP[NLEV];
  const float* boxP[NLEV];
  const float* ctrP[NLEV];
  for (int l = 0; l < NLEV; ++l) {
    clsP[l] = (const float*)d_in[3 * l + 0];
    boxP[l] = (const float*)d_in[3 * l + 1];
    ctrP[l] = (const float*)d_in[3 * l + 2];
  }
  int cumP[NLEV + 1];
  cumP[0] = 0;
  for (int l = 0; l < NLEV; ++l) cumP[l + 1] = cumP[l] + LP[l];
  const int totP = cumP[NLEV];  // 20267

  // carve workspace (total ~10 MB)
  uintptr_t w = (uintptr_t)d_ws;
  auto carve = [&](size_t bytes) -> uintptr_t {
    uintptr_t p = w;
    w += (bytes + 255) & ~(size_t)255;
    return p;
  };
  float* ctrSig = (float*)carve((size_t)NIMG * totP * 4);
  u32*   hist   = (u32*)carve((size_t)NIMG * NLEV * NBIN * 4);
  u32*   cnt    = (u32*)carve((size_t)NIMG * NLEV * 4);
  int*   thr    = (int*)carve((size_t)NIMG * NLEV * 4);
  u64*   keys   = (u64*)carve((size_t)NIMG * NLEV * CAP * 8);
  float* cand   = (float*)carve((size_t)NIMG * CANDP * CSTRIDE * 4);
  float* sorted = (float*)carve((size_t)NIMG * CANDP * CSTRIDE * 4);

  // zero atomic state every call (deterministic across graph replays)
  {
    int nz = NIMG * NLEV * NBIN;
    k_zero_u32<<<(nz + 255) / 256, 256, 0, stream>>>(hist, nz);
    k_zero_u32<<<1, 128, 0, stream>>>(cnt, NIMG * NLEV);
  }
  for (int l = 0; l < NLEV; ++l) {
    int n = NIMG * LP[l];
    k_ctr_sigmoid<<<(n + 255) / 256, 256, 0, stream>>>(ctrP[l], ctrSig + (size_t)NIMG * cumP[l], n);
  }
  for (int l = 0; l < NLEV; ++l) {
    dim3 grid((LP[l] + 255) / 256, NIMG * NC);
    k_hist<<<grid, 256, 0, stream>>>(clsP[l], ctrSig + (size_t)NIMG * cumP[l], hist, LP[l], l);
  }
  k_thresh<<<1, 128, 0, stream>>>(hist, thr);
  for (int l = 0; l < NLEV; ++l) {
    dim3 grid((LP[l] + 255) / 256, NIMG * NC);
    k_compact<<<grid, 256, 0, stream>>>(clsP[l], ctrSig + (size_t)NIMG * cumP[l],
                                        thr, cnt, keys, LP[l], l);
  }
  LevelMeta LM;
  for (int l = 0; l < NLEV; ++l) {
    LM.box[l] = boxP[l];
    LM.P[l] = LP[l];
    LM.W[l] = LW_[l];
    LM.S[l] = LS_[l];
  }
  k_level_topk<<<NIMG * NLEV, 512, 0, stream>>>(cnt, keys, LM, cand);
  k_image_sort<<<NIMG, 1024, 0, stream>>>(cand, sorted);
  k_nms<<<NIMG, 1024, 0, stream>>>(sorted, (float*)d_out);
}